// EdgeModel_out_31748398252727
// MI455X (gfx1250) — compile-verified
//
#include <hip/hip_runtime.h>

typedef float v2f __attribute__((ext_vector_type(2)));
typedef float v8f __attribute__((ext_vector_type(8)));

__device__ __forceinline__ v2f ldf2(const float* p) {
  return *reinterpret_cast<const v2f*>(p);
}

// One wave (32 lanes) computes 16 edges.
// A-matrix (16x4 f32): lane L holds row M=L%16, K-pair (2*(L/16), 2*(L/16)+1).
// B-matrix (4x16 f32): W1 chunk broadcast into all 16 columns; same lane-half skew.
// D (16x16 f32): lane 0 holds M=0..7 in its 8 VGPRs, lane 16 holds M=8..15.
__global__ __launch_bounds__(256)
void edge_model_wmma(const float* __restrict__ x_h,
                     const float* __restrict__ x_g,
                     const float* __restrict__ edge_attr,
                     const float* __restrict__ u,
                     const int*   __restrict__ edge_index,
                     const int*   __restrict__ batch_e,
                     const float* __restrict__ W1,
                     const float* __restrict__ b1,
                     const float* __restrict__ W2,
                     const float* __restrict__ b2,
                     float* __restrict__ out,
                     int nEdges, int nTiles)
{
  const int wavesPerBlock = blockDim.x >> 5;
  const int tile = blockIdx.x * wavesPerBlock + (int)(threadIdx.x >> 5);
  if (tile >= nTiles) return;              // wave-uniform: EXEC stays all-1s
  const int lane = threadIdx.x & 31;
  const int m  = lane & 15;                // edge row within tile
  const int kh = lane >> 4;                // K half: 0 -> K{0,1}, 1 -> K{2,3}
  const int e  = tile * 16 + m;

  const int src = edge_index[e];           // row 0 of edge_index
  const int tgt = edge_index[nEdges + e];  // row 1 of edge_index
  const int g   = batch_e[e];

  const float* ph = x_h       + (size_t)src * 64 + 2 * kh;
  const float* pg = x_g       + (size_t)tgt * 64 + 2 * kh;
  const float* pe = edge_attr + (size_t)e   * 8  + 2 * kh;
  const float* pu = u         + (size_t)g   * 16 + 2 * kh;
  const float* pw = W1 + 2 * kh;

  v8f c0 = {}; v8f c1 = {};                // two accumulators for ILP

#define STEP(PTR, WIDX, ACC) {                                              \
    v2f a  = ldf2(PTR);                                                     \
    v2f bv = ldf2(pw + (WIDX));                                             \
    ACC = __builtin_amdgcn_wmma_f32_16x16x4_f32(                            \
        false, a, false, bv, (short)0, ACC, false, false); }

  // x_h segment: K = 0..63  (16 steps of K=4)
#pragma unroll
  for (int s = 0; s < 8; ++s) {
    STEP(ph + 8 * s,     8 * s,     c0);
    STEP(ph + 8 * s + 4, 8 * s + 4, c1);
  }
  // x_g segment: K = 64..127
#pragma unroll
  for (int s = 0; s < 8; ++s) {
    STEP(pg + 8 * s,     64 + 8 * s,     c0);
    STEP(pg + 8 * s + 4, 64 + 8 * s + 4, c1);
  }
  // edge_attr segment: K = 128..135 (2 steps)
  STEP(pe,     128, c0);
  STEP(pe + 4, 132, c1);
  // u segment: K = 136..151 (4 steps)
#pragma unroll
  for (int s = 0; s < 2; ++s) {
    STEP(pu + 8 * s,     136 + 8 * s,     c0);
    STEP(pu + 8 * s + 4, 136 + 8 * s + 4, c1);
  }
#undef STEP

  const float bias1 = b1[0], w2 = W2[0], bias2 = b2[0];
  v8f r;
#pragma unroll
  for (int i = 0; i < 8; ++i) {
    float h = c0[i] + c1[i] + bias1;       // dot(feat, W1) + b1
    h = (h >= 0.0f) ? h : 0.1f * h;        // LeakyReLU(0.1)
    r[i] = h * w2 + bias2;                 // @ W2 + b2
  }

  // lane 0 -> edges 0..7, lane 16 -> edges 8..15 (two b128 stores each)
  if (m == 0) {
    float* po = out + (size_t)tile * 16 + kh * 8;
    reinterpret_cast<float4*>(po)[0] = make_float4(r[0], r[1], r[2], r[3]);
    reinterpret_cast<float4*>(po)[1] = make_float4(r[4], r[5], r[6], r[7]);
  }
}

// Scalar tail for E % 16 != 0 (not hit for E = 1,000,000 but kept generic).
__global__ void edge_model_tail(const float* __restrict__ x_h,
                                const float* __restrict__ x_g,
                                const float* __restrict__ edge_attr,
                                const float* __restrict__ u,
                                const int*   __restrict__ edge_index,
                                const int*   __restrict__ batch_e,
                                const float* __restrict__ W1,
                                const float* __restrict__ b1,
                                const float* __restrict__ W2,
                                const float* __restrict__ b2,
                                float* __restrict__ out,
                                int start, int nEdges)
{
  int e = start + blockIdx.x * blockDim.x + threadIdx.x;
  if (e >= nEdges) return;
  const int src = edge_index[e];
  const int tgt = edge_index[nEdges + e];
  const int g   = batch_e[e];
  float acc = 0.0f;
  for (int i = 0; i < 64; ++i) acc += x_h[(size_t)src * 64 + i] * W1[i];
  for (int i = 0; i < 64; ++i) acc += x_g[(size_t)tgt * 64 + i] * W1[64 + i];
  for (int i = 0; i < 8;  ++i) acc += edge_attr[(size_t)e * 8 + i] * W1[128 + i];
  for (int i = 0; i < 16; ++i) acc += u[(size_t)g * 16 + i] * W1[136 + i];
  float h = acc + b1[0];
  h = (h >= 0.0f) ? h : 0.1f * h;
  out[e] = h * W2[0] + b2[0];
}

extern "C" void kernel_launch(void* const* d_in, const int* in_sizes, int n_in,
                              void* d_out, int out_size, void* d_ws, size_t ws_size,
                              hipStream_t stream) {
  const float* x_h       = (const float*)d_in[0];
  const float* x_g       = (const float*)d_in[1];
  const float* edge_attr = (const float*)d_in[2];
  const float* u         = (const float*)d_in[3];
  const int*   edge_idx  = (const int*)  d_in[4];
  const int*   batch_e   = (const int*)  d_in[5];
  const float* W1        = (const float*)d_in[6];
  const float* b1        = (const float*)d_in[7];
  const float* W2        = (const float*)d_in[8];
  const float* b2        = (const float*)d_in[9];
  float* out = (float*)d_out;

  const int nEdges = in_sizes[5];          // batch_e has one entry per edge
  const int nTiles = nEdges / 16;
  const int rem    = nEdges - nTiles * 16;

  if (nTiles > 0) {
    const int wavesPerBlock = 8;           // 256 threads = 8 wave32
    const int blocks = (nTiles + wavesPerBlock - 1) / wavesPerBlock;
    edge_model_wmma<<<blocks, 256, 0, stream>>>(
        x_h, x_g, edge_attr, u, edge_idx, batch_e,
        W1, b1, W2, b2, out, nEdges, nTiles);
  }
  if (rem > 0) {
    edge_model_tail<<<(rem + 63) / 64, 64, 0, stream>>>(
        x_h, x_g, edge_attr, u, edge_idx, batch_e,
        W1, b1, W2, b2, out, nTiles * 16, nEdges);
  }
}